// KernelizedAttentionResBlock_71528385347626
// MI455X (gfx1250) — compile-verified
//
#include <hip/hip_runtime.h>
#include <hip/hip_bf16.h>
#include <math.h>

typedef __attribute__((ext_vector_type(2))) float v2f;
typedef __attribute__((ext_vector_type(8))) float v8f;

#define LN_EPS 1e-5f
#define N_TOK 1024

// ---------------------------------------------------------------------------
// LayerNorm over last dim of a (32, 1024) matrix. One block per row.
// ---------------------------------------------------------------------------
__global__ void ln_kernel(const float* __restrict__ X,
                          const float* __restrict__ gamma,
                          const float* __restrict__ beta,
                          float* __restrict__ Y) {
    __shared__ float s_sum[256];
    __shared__ float s_sq[256];
    const int b = blockIdx.x;
    const float* x = X + (size_t)b * N_TOK;
    float sum = 0.f, sq = 0.f;
    for (int i = threadIdx.x; i < N_TOK; i += blockDim.x) {
        float v = x[i];
        sum += v;
        sq  += v * v;
    }
    s_sum[threadIdx.x] = sum;
    s_sq[threadIdx.x]  = sq;
    __syncthreads();
    for (int off = 128; off > 0; off >>= 1) {
        if ((int)threadIdx.x < off) {
            s_sum[threadIdx.x] += s_sum[threadIdx.x + off];
            s_sq[threadIdx.x]  += s_sq[threadIdx.x + off];
        }
        __syncthreads();
    }
    const float mu  = s_sum[0] * (1.0f / N_TOK);
    const float var = s_sq[0] * (1.0f / N_TOK) - mu * mu;
    const float inv = rsqrtf(var + LN_EPS);
    float* y = Y + (size_t)b * N_TOK;
    for (int i = threadIdx.x; i < N_TOK; i += blockDim.x) {
        y[i] = (x[i] - mu) * inv * gamma[i] + beta[i];
    }
}

// ---------------------------------------------------------------------------
// out[m][n] = act( sum_k A[m][k] * W[n][k] + bias[n] )   (+ residual for ACT=3)
// M fixed = 32 (two 16-row tiles). One wave (32 threads) per 16x16 tile,
// fp32 WMMA 16x16x4: A tile = 2 VGPRs (lanes 0-15: K=k0,k0+1; lanes 16-31:
// K=k0+2,k0+3), B mirrored.
// ACT: 0 = identity, 1 = tanh, 2 = silu, 3 = identity + residual add
// ---------------------------------------------------------------------------
template <int ACT>
__global__ void gemm_wmma_kernel(const float* __restrict__ A,    // (32, K)
                                 const float* __restrict__ W,    // (N, K) row-major
                                 const float* __restrict__ bias, // (N)
                                 const float* __restrict__ res,  // (32, N) or null
                                 float* __restrict__ out,        // (32, N)
                                 int K, int N) {
    const int tileN = blockIdx.x;          // N / 16 tiles
    const int tileM = blockIdx.y;          // 0 or 1
    const int lane  = threadIdx.x & 31;
    const int half  = lane >> 4;           // 0: K pair {0,1}; 1: K pair {2,3}
    const int l16   = lane & 15;

    const float* ap = A + (size_t)(tileM * 16 + l16) * K + half * 2;
    const float* wp = W + (size_t)(tileN * 16 + l16) * K + half * 2;

    v8f c = {};
    for (int k0 = 0; k0 < K; k0 += 4) {
        v2f a = *(const v2f*)(ap + k0);
        v2f b = *(const v2f*)(wp + k0);
        __builtin_prefetch(wp + k0 + 256, 0, 1);   // global_prefetch_b8 on weights
        c = __builtin_amdgcn_wmma_f32_16x16x4_f32(
                /*neg_a=*/false, a, /*neg_b=*/false, b,
                /*c_mod=*/(short)0, c, /*reuse_a=*/false, /*reuse_b=*/false);
    }

    const int n  = tileN * 16 + l16;
    const float bn = bias[n];
    const int m0 = tileM * 16 + half * 8;  // C layout: lanes>=16 hold rows M+8..M+15
#pragma unroll
    for (int r = 0; r < 8; ++r) {
        float v = c[r] + bn;
        if (ACT == 1) v = tanhf(v);
        else if (ACT == 2) v = v / (1.f + __expf(-v));   // silu
        const int m = m0 + r;
        if (ACT == 3) v += res[(size_t)m * N + n];
        out[(size_t)m * N + n] = v;
    }
}

// ---------------------------------------------------------------------------
// x[b][i] = Q[b][i] + sum_d exp(-0.5*(K[b][i][d]-mu[b][i])^2/(sigma^2+1e-8)) * V[b][i][d]
// One wave per (b,i) row; float4 streaming loads of K and V (256 MB total —
// this kernel is the HBM-bound bulk of the op).
// ---------------------------------------------------------------------------
__global__ void attn_kernel(const float* __restrict__ Kmat,
                            const float* __restrict__ Vmat,
                            const float* __restrict__ Q,
                            const float* __restrict__ mu,
                            const float* __restrict__ sigma,
                            float* __restrict__ X) {
    const int wave = threadIdx.x >> 5;
    const int lane = threadIdx.x & 31;
    const int row  = blockIdx.x * 8 + wave;       // 0 .. 32*1024-1

    const float m   = mu[row];
    const float s   = sigma[row];
    const float inv = -0.5f / (s * s + 1e-8f);

    const float4* kp = (const float4*)(Kmat + (size_t)row * 1024);
    const float4* vp = (const float4*)(Vmat + (size_t)row * 1024);

    float sum = 0.f;
    for (int it = lane; it < 256; it += 32) {     // 1024 elems / 4 per lane
        float4 k4 = kp[it];
        float4 v4 = vp[it];
        float d0 = k4.x - m, d1 = k4.y - m, d2 = k4.z - m, d3 = k4.w - m;
        sum += __expf(d0 * d0 * inv) * v4.x;
        sum += __expf(d1 * d1 * inv) * v4.y;
        sum += __expf(d2 * d2 * inv) * v4.z;
        sum += __expf(d3 * d3 * inv) * v4.w;
    }
    // wave32 reduction
    for (int off = 16; off > 0; off >>= 1)
        sum += __shfl_xor(sum, off, 32);
    if (lane == 0)
        X[row] = sum + Q[row];
}

// ---------------------------------------------------------------------------
extern "C" void kernel_launch(void* const* d_in, const int* in_sizes, int n_in,
                              void* d_out, int out_size, void* d_ws, size_t ws_size,
                              hipStream_t stream) {
    const float* Q       = (const float*)d_in[0];
    const float* Kmat    = (const float*)d_in[1];
    const float* Vmat    = (const float*)d_in[2];
    const float* mu_w    = (const float*)d_in[3];
    const float* mu_b    = (const float*)d_in[4];
    const float* sigma_w = (const float*)d_in[5];
    const float* sigma_b = (const float*)d_in[6];
    const float* ffn_w1  = (const float*)d_in[7];
    const float* ffn_b1  = (const float*)d_in[8];
    const float* ffn_w2  = (const float*)d_in[9];
    const float* ffn_b2  = (const float*)d_in[10];
    const float* ln_ff_g = (const float*)d_in[11];
    const float* ln_ff_b = (const float*)d_in[12];
    const float* ln_q_g  = (const float*)d_in[13];
    const float* ln_q_b  = (const float*)d_in[14];

    const int Bb = 32, n = 1024, mff = 4096;

    float* q     = (float*)d_ws;          // (32, 1024)
    float* mu    = q     + Bb * n;        // (32, 1024)
    float* sigma = mu    + Bb * n;        // (32, 1024)
    float* x     = sigma + Bb * n;        // (32, 1024)
    float* h     = x     + Bb * n;        // (32, 1024)
    float* t     = h     + Bb * n;        // (32, 4096)
    float* out   = (float*)d_out;

    // 1. q = LN(Q)
    ln_kernel<<<Bb, 256, 0, stream>>>(Q, ln_q_g, ln_q_b, q);

    // 2. mu = tanh(q @ mu_w.T + mu_b);  sigma = q @ sigma_w.T + sigma_b
    gemm_wmma_kernel<1><<<dim3(n / 16, 2), 32, 0, stream>>>(q, mu_w, mu_b, nullptr, mu, n, n);
    gemm_wmma_kernel<0><<<dim3(n / 16, 2), 32, 0, stream>>>(q, sigma_w, sigma_b, nullptr, sigma, n, n);

    // 3. x = Q + sum_d exp(...) * V   (streams K and V: the HBM-bound bulk)
    attn_kernel<<<(Bb * n) / 8, 256, 0, stream>>>(Kmat, Vmat, Q, mu, sigma, x);

    // 4. h = LN(x)
    ln_kernel<<<Bb, 256, 0, stream>>>(x, ln_ff_g, ln_ff_b, h);

    // 5. t = silu(h @ ffn_w1.T + ffn_b1)
    gemm_wmma_kernel<2><<<dim3(mff / 16, 2), 32, 0, stream>>>(h, ffn_w1, ffn_b1, nullptr, t, n, mff);

    // 6. out = x + (t @ ffn_w2.T + ffn_b2)
    gemm_wmma_kernel<3><<<dim3(n / 16, 2), 32, 0, stream>>>(t, ffn_w2, ffn_b2, x, out, mff, n);
}